// GaussianMultiScore_44951127720630
// MI455X (gfx1250) — compile-verified
//
#include <hip/hip_runtime.h>
#include <stdint.h>

// GaussianMultiScore for MI455X (gfx1250, wave32).
//
// Roofline: ~580 B/item streamed, ~600 FLOP/item -> AI ~ 1 FLOP/B.
// At 23.3 TB/s the floor is ~13 us; VALU issue is ~2 us -> memory-bound.
// CDNA5 paths: global_load_async_to_lds_b128 tile staging (ASYNCcnt) +
// s_wait_asynccnt, padded LDS layout (272B slots, bank-conflict-free b128
// reads), wave32 blocks. WMMA deliberately NOT used: the math is batched
// per-lane 8x8 (block-diagonal), which cannot map onto wave-wide 16x16
// D=A*B without cross-lane marshalling that costs more than it saves.
//
// Algebra: with the reverse (upper) Cholesky lam = U*U^T,
//   chol(inv(lam)) = U^{-T}  =>  diag_new[i] = 1/U[i][i]
//   mu_new = U^{-T} U^{-1} eta          (two triangular solves)
//   diff^T lam diff = ||U^T diff||^2    (reference var_inv == lam)
//   sum log(diag_new^2) = -log(prod U[k][k]^2)
// so one factorization replaces inverse + two Choleskys + another inverse,
// and all exp/log terms fuse into a single expf.

#define DD     8
#define ITEMS  128           // items per block == threads per block (4 wave32)
#define PADF   68            // floats per matrix slot: 64 + 4 pad = 272 B
#define LOG2PI 1.83787706640934548356f

__global__ __launch_bounds__(ITEMS)
void gms_kernel(const float* __restrict__ mu0, const float* __restrict__ mu1,
                const float* __restrict__ cho0, const float* __restrict__ cho1,
                const float* __restrict__ sample, float* __restrict__ out)
{
    __shared__ __align__(16) float smem[2 * ITEMS * PADF];   // 69,632 bytes
    const int tid = threadIdx.x;
    const size_t item0 = (size_t)blockIdx.x * ITEMS;

    // ---- Stage cho0/cho1 tiles into LDS via CDNA5 async global->LDS copies ----
    // Chunk c = tid + 128k -> item = (tid>>4) + 8k, intra = (tid&15)*16B.
    // Both LDS (+2176B) and global (+2048B) offsets are linear in k.
    {
        const uint32_t lbase =
            (uint32_t)(uintptr_t)smem + (uint32_t)(((tid >> 4) * PADF + ((tid & 15) << 2)) << 2);
        const char* gb0 = (const char*)(cho0 + item0 * 64) + (size_t)tid * 16;
        const char* gb1 = (const char*)(cho1 + item0 * 64) + (size_t)tid * 16;
        #pragma unroll
        for (int k = 0; k < 16; ++k) {
            const uint32_t l0 = lbase + (uint32_t)(k * 8 * PADF * 4);       // +2176/iter
            const uint32_t l1 = l0 + (uint32_t)(ITEMS * PADF * 4);          // cho1 region
            const char* p0 = gb0 + (size_t)k * 2048;                        // 512B/wave-instr
            const char* p1 = gb1 + (size_t)k * 2048;
            asm volatile("global_load_async_to_lds_b128 %0, %1, off"
                         :: "v"(l0), "v"(p0) : "memory");
            asm volatile("global_load_async_to_lds_b128 %0, %1, off"
                         :: "v"(l1), "v"(p1) : "memory");
        }
    }

    // ---- mu / sample loads while async copies are in flight ----
    float m0[DD], m1[DD], smp[DD];
    {
        const float4* v0 = reinterpret_cast<const float4*>(mu0) + (item0 + tid) * 2;
        const float4* v1 = reinterpret_cast<const float4*>(mu1) + (item0 + tid) * 2;
        float4 a = v0[0], b = v0[1], c = v1[0], d = v1[1];
        m0[0]=a.x; m0[1]=a.y; m0[2]=a.z; m0[3]=a.w; m0[4]=b.x; m0[5]=b.y; m0[6]=b.z; m0[7]=b.w;
        m1[0]=c.x; m1[1]=c.y; m1[2]=c.z; m1[3]=c.w; m1[4]=d.x; m1[5]=d.y; m1[6]=d.z; m1[7]=d.w;
        #pragma unroll
        for (int i = 0; i < DD; ++i) smp[i] = sample[i];   // uniform -> s_load
    }

    asm volatile("s_wait_asynccnt 0" ::: "memory");
    __syncthreads();

    const float* A0 = &smem[tid * PADF];
    const float* A1 = &smem[ITEMS * PADF + tid * PADF];

    // ---- eta0/eta1, quadratic terms, diag products, lam = cho0 + cho1 ----
    float lam[DD][DD], eta0[DD], eta1[DD];
    float P0 = 1.f, P1 = 1.f;
    #pragma unroll
    for (int i = 0; i < DD; ++i) {
        float e0 = 0.f, e1 = 0.f;
        #pragma unroll
        for (int j = 0; j < DD; ++j) {
            const float r0 = A0[i * DD + j];
            const float r1 = A1[i * DD + j];
            e0 += r0 * m0[j];
            e1 += r1 * m1[j];
            lam[i][j] = r0 + r1;
            if (j == i) { P0 *= r0; P1 *= r1; }
        }
        eta0[i] = e0; eta1[i] = e1;
    }
    float q0 = 0.f, q1 = 0.f, eta[DD];
    #pragma unroll
    for (int i = 0; i < DD; ++i) {
        q0 += m0[i] * eta0[i];
        q1 += m1[i] * eta1[i];
        eta[i] = eta0[i] + eta1[i];
    }

    // ---- Reverse (upper) Cholesky in place: lam = U * U^T, U upper ----
    float rinv[DD];
    #pragma unroll
    for (int k = DD - 1; k >= 0; --k) {
        float s = lam[k][k];
        #pragma unroll
        for (int j = k + 1; j < DD; ++j) s -= lam[k][j] * lam[k][j];
        const float ri = rsqrtf(s);        // v_rsq_f32: gives both 1/U[k][k] ...
        rinv[k] = ri;
        lam[k][k] = s * ri;                // ... and U[k][k] = s * rsqrt(s)
        #pragma unroll
        for (int i = 0; i < k; ++i) {
            float t = lam[i][k];
            #pragma unroll
            for (int j = k + 1; j < DD; ++j) t -= lam[i][j] * lam[k][j];
            lam[i][k] = t * ri;
        }
    }

    // ---- mu_new = U^{-T} U^{-1} eta  (two triangular solves) ----
    float y[DD];
    #pragma unroll
    for (int i = DD - 1; i >= 0; --i) {            // U y = eta (back-subst)
        float t = eta[i];
        #pragma unroll
        for (int j = i + 1; j < DD; ++j) t -= lam[i][j] * y[j];
        y[i] = t * rinv[i];
    }
    float x[DD];
    #pragma unroll
    for (int i = 0; i < DD; ++i) {                 // U^T x = y (fwd-subst)
        float t = y[i];
        #pragma unroll
        for (int j = 0; j < i; ++j) t -= lam[j][i] * x[j];
        x[i] = t * rinv[i];
    }

    float qN = 0.f, diff[DD];
    #pragma unroll
    for (int i = 0; i < DD; ++i) { qN += x[i] * eta[i]; diff[i] = x[i] - smp[i]; }

    // diff^T lam diff = || U^T diff ||^2
    float qd = 0.f;
    #pragma unroll
    for (int k = 0; k < DD; ++k) {
        float t = 0.f;
        #pragma unroll
        for (int i = 0; i <= k; ++i) t += lam[i][k] * diff[i];
        qd += t * t;
    }

    // det (faithful quirk: SUM of diag_new^2); merged log terms:
    //   -s0 - s1 + sN = -log((P0*P1*PU)^2)
    float PU = 1.f, det = 0.f;
    #pragma unroll
    for (int k = 0; k < DD; ++k) { PU *= lam[k][k]; det += rinv[k] * rinv[k]; }
    const float PP = P0 * P1 * PU;                 // in [1, ~2e4]: safe in f32

    const float E = -0.5f * (16.f * LOG2PI + q0 + q1 - qN + qd
                             + logf(det) - logf(PP * PP));
    out[item0 + tid] = expf(E);
}

extern "C" void kernel_launch(void* const* d_in, const int* in_sizes, int n_in,
                              void* d_out, int out_size, void* d_ws, size_t ws_size,
                              hipStream_t stream) {
    const float* mu0    = (const float*)d_in[0];
    const float* mu1    = (const float*)d_in[1];
    const float* cho0   = (const float*)d_in[2];
    const float* cho1   = (const float*)d_in[3];
    const float* sample = (const float*)d_in[4];
    float* out = (float*)d_out;

    const int Bn   = in_sizes[0] / DD;     // 524288
    const int grid = Bn / ITEMS;           // 4096 blocks of 128 threads (4 wave32)
    hipLaunchKernelGGL(gms_kernel, dim3(grid), dim3(ITEMS), 0, stream,
                       mu0, mu1, cho0, cho1, sample, out);
}